// tMGRU_20048907338171
// MI455X (gfx1250) — compile-verified
//
#include <hip/hip_runtime.h>
#include <hip/hip_bf16.h>

typedef __attribute__((ext_vector_type(2))) float v2f;
typedef __attribute__((ext_vector_type(8))) float v8f;

#define HS 1024
#define DVARS 33
#define NTHREADS 192   // 6 waves of 32

__global__ __launch_bounds__(NTHREADS)
void tmgru_fused_wmma(const float* __restrict__ H,   const float* __restrict__ x,
                      const float* __restrict__ h,
                      const float* __restrict__ W_z, const float* __restrict__ U_zx,
                      const float* __restrict__ U_zh,
                      const float* __restrict__ W_r, const float* __restrict__ U_rx,
                      const float* __restrict__ U_rh,
                      const float* __restrict__ W_h, const float* __restrict__ U_hx,
                      const float* __restrict__ U_hh,
                      const float* __restrict__ b_z, const float* __restrict__ b_r,
                      const float* __restrict__ b_h,
                      float* __restrict__ out)
{
    __shared__ float s_vec[2][HS];   // [0] = H[d], [1] = h
    __shared__ float s_acc[6][16];   // 16-row GEMV result per matrix

    const int d    = blockIdx.y;          // variable index
    const int row0 = blockIdx.x * 16;     // output row tile
    const int tid  = threadIdx.x;
    const int wave = tid >> 5;
    const int lane = tid & 31;

    // Stage the two multiplied vectors into LDS once per block.
    for (int i = tid; i < HS; i += NTHREADS) {
        s_vec[0][i] = H[d * HS + i];
        s_vec[1][i] = h[i];
    }
    __syncthreads();

    // Wave w computes dot(W_w[row0..row0+15, :], v_w) via fp32 WMMA.
    {
        const float* Wmat; int vsel;
        switch (wave) {
            case 0:  Wmat = W_z;  vsel = 0; break;
            case 1:  Wmat = U_zh; vsel = 1; break;
            case 2:  Wmat = W_r;  vsel = 0; break;
            case 3:  Wmat = U_rh; vsel = 1; break;
            case 4:  Wmat = W_h;  vsel = 0; break;
            default: Wmat = U_hh; vsel = 1; break;
        }
        const int mrow = lane & 15;          // M row within the 16-row tile
        const int koff = (lane >> 4) << 1;   // K pair: 0 for lanes 0-15, 2 for 16-31

        const float* arow = Wmat + (size_t)d * HS * HS + (size_t)(row0 + mrow) * HS + koff;
        const float* bvec = &s_vec[vsel][koff];

        v8f acc0 = {};
        v8f acc1 = {};
        #pragma unroll 4
        for (int k = 0; k < HS; k += 8) {
            __builtin_prefetch(arow + k + 64, 0, 0);   // global_prefetch_b8, ~256B ahead
            v2f a0 = *(const v2f*)(arow + k);          // A frag (global_load_b64)
            v2f b0 = *(const v2f*)(bvec + k);          // B frag broadcast (ds_load_b64)
            v2f a1 = *(const v2f*)(arow + k + 4);
            v2f b1 = *(const v2f*)(bvec + k + 4);
            // 8 args: (neg_a, A, neg_b, B, c_mod, C, reuse_a, reuse_b)
            acc0 = __builtin_amdgcn_wmma_f32_16x16x4_f32(false, a0, false, b0,
                                                         (short)0, acc0, false, false);
            acc1 = __builtin_amdgcn_wmma_f32_16x16x4_f32(false, a1, false, b1,
                                                         (short)0, acc1, false, false);
        }
        v8f acc = acc0 + acc1;

        // All N columns of D are identical (B broadcast). C/D layout:
        // lane n (0-15): VGPR g holds row row0+g; lanes 16-31: row row0+8+g.
        if (lane == 0) {
            #pragma unroll
            for (int g = 0; g < 8; ++g) s_acc[wave][g] = acc[g];
        } else if (lane == 16) {
            #pragma unroll
            for (int g = 0; g < 8; ++g) s_acc[wave][8 + g] = acc[g];
        }
    }
    __syncthreads();

    // Elementwise gate math for the 16 rows of this tile.
    if (tid < 16) {
        const int i  = row0 + tid;
        const float xd = x[d];
        const float wzH  = s_acc[0][tid], uzh = s_acc[1][tid];
        const float wrH  = s_acc[2][tid], urh = s_acc[3][tid];
        const float whH  = s_acc[4][tid], uhh = s_acc[5][tid];

        const float z_pre = wzH + uzh + U_zx[d * HS + i] * xd + b_z[i];
        const float r_pre = wrH + urh + U_rx[d * HS + i] * xd + b_r[i];
        const float z_t = 1.0f / (1.0f + __expf(-z_pre));
        const float r_t = 1.0f / (1.0f + __expf(-r_pre));
        // r_t gates ONLY the W_h @ H term (matches reference).
        const float H_hat = tanhf(r_t * whH + U_hx[d * HS + i] * xd + uhh + b_h[i]);
        const float Hi = H[d * HS + i];
        out[d * HS + i] = z_t * Hi + (1.0f - z_t) * H_hat;
    }
}

extern "C" void kernel_launch(void* const* d_in, const int* in_sizes, int n_in,
                              void* d_out, int out_size, void* d_ws, size_t ws_size,
                              hipStream_t stream) {
    (void)in_sizes; (void)n_in; (void)d_ws; (void)ws_size; (void)out_size;
    const float* H    = (const float*)d_in[0];
    const float* x    = (const float*)d_in[1];
    const float* h    = (const float*)d_in[2];
    const float* W_z  = (const float*)d_in[3];
    const float* U_zx = (const float*)d_in[4];
    const float* U_zh = (const float*)d_in[5];
    const float* W_r  = (const float*)d_in[6];
    const float* U_rx = (const float*)d_in[7];
    const float* U_rh = (const float*)d_in[8];
    const float* W_h  = (const float*)d_in[9];
    const float* U_hx = (const float*)d_in[10];
    const float* U_hh = (const float*)d_in[11];
    const float* b_z  = (const float*)d_in[12];
    const float* b_r  = (const float*)d_in[13];
    const float* b_h  = (const float*)d_in[14];
    float* out = (float*)d_out;

    dim3 grid(HS / 16, DVARS);   // 64 row-tiles x 33 variables
    dim3 block(NTHREADS);
    tmgru_fused_wmma<<<grid, block, 0, stream>>>(H, x, h,
                                                 W_z, U_zx, U_zh,
                                                 W_r, U_rx, U_rh,
                                                 W_h, U_hx, U_hh,
                                                 b_z, b_r, b_h, out);
}